// TTEmbeddingBag_41308995453043
// MI455X (gfx1250) — compile-verified
//
#include <hip/hip_runtime.h>

typedef __attribute__((ext_vector_type(2))) float v2f;
typedef __attribute__((ext_vector_type(8))) float v8f;

#define TT_P0 100
#define TT_P1 100
#define TT_P2 100
#define TT_R1 32
#define TT_R2 32
#define TT_D  128
#define AB_ROW 512            // 16 (q1*q2) x 32 (r2) floats per (i1,i2) pair
#define NPAIR (TT_P0 * TT_P1)

// ---------------- bag offsets: exclusive scan of lengths ----------------
__global__ void tt_scan_offsets(const int* __restrict__ lengths,
                                int* __restrict__ offsets, int B) {
  __shared__ int part[256];
  int t = threadIdx.x;
  int per = (B + 255) >> 8;
  int lo = t * per, hi = lo + per;
  if (lo > B) lo = B;
  if (hi > B) hi = B;
  int s = 0;
  for (int i = lo; i < hi; ++i) s += lengths[i];
  part[t] = s;
  __syncthreads();
  for (int off = 1; off < 256; off <<= 1) {
    int add = (t >= off) ? part[t - off] : 0;
    __syncthreads();
    part[t] += add;
    __syncthreads();
  }
  int base = (t == 0) ? 0 : part[t - 1];
  for (int i = lo; i < hi; ++i) { offsets[i] = base; base += lengths[i]; }
}

// ---------------- Phase A: AB[i1,i2] = core0[i1] @ core1[i2] ----------------
// For a block of 4 p1 rows (M = 4 p1 * 4 q1 = 16) and one p2:
//   A[16,32] (K = r1), B[32,128] (N = j = q2*32+r2), 8 column tiles of 16.
// One wave per column tile; 8 waves per block; grid = 25 * 100 blocks.
__global__ void __launch_bounds__(256)
tt_phaseA(const float* __restrict__ core0,
          const float* __restrict__ core1,
          float* __restrict__ AB) {
  int blk  = blockIdx.x;
  int p1b  = blk / TT_P1;           // 0..24 (block of 4 p1 values)
  int p2   = blk % TT_P1;
  int lane = threadIdx.x & 31;
  int ntile = threadIdx.x >> 5;     // 0..7
  int ml   = lane & 15;             // A/B row-or-col index within half-wave
  int hi2  = (lane >> 4) << 1;      // K sub-offset: 0 (lanes 0-15) / 2 (16-31)

  const float* arow  = core0 + (size_t)(p1b * 4 + (ml >> 2)) * 128 + (ml & 3) * 32;
  const float* bbase = core1 + (size_t)p2 * 4096 + ntile * 16 + ml;

  v8f acc = {};
  for (int kk = 0; kk < 8; ++kk) {
    int k = kk * 4 + hi2;
    v2f A;  A.x  = arow[k];                 A.y  = arow[k + 1];
    v2f Bv; Bv.x = bbase[(size_t)k * 128];  Bv.y = bbase[(size_t)(k + 1) * 128];
    acc = __builtin_amdgcn_wmma_f32_16x16x4_f32(false, A, false, Bv,
                                                (short)0, acc, false, false);
  }
#pragma unroll
  for (int v = 0; v < 8; ++v) {
    int M  = v + ((lane >> 4) << 3);        // 0..15 output row
    int p1 = p1b * 4 + (M >> 2);
    int q1 = M & 3;
    AB[(size_t)(p1 * TT_P1 + p2) * AB_ROW + q1 * 128 + ntile * 16 + ml] = acc[v];
  }
}

// ---------------- Phase B: per-bag stage-2 GEMM + pooled accumulate --------
// One wave per bag. Two indices per WMMA chain: A = [AB_n1 | AB_n2] (16x64),
// B = blockdiag(c_n1, c_n2) (64x16). Bag sum lives in the C accumulator.
__global__ void __launch_bounds__(256)
tt_phaseB(const int* __restrict__ indices,
          const int* __restrict__ lengths,
          const int* __restrict__ offsets,
          const float* __restrict__ AB,
          const float* __restrict__ core2,
          float* __restrict__ out, int B) {
  int wave = threadIdx.x >> 5;
  int bag  = blockIdx.x * 8 + wave;
  if (bag >= B) return;
  int lane = threadIdx.x & 31;
  int ml   = lane & 15;
  int hi2  = (lane >> 4) << 1;

  int start = offsets[bag];
  int len   = lengths[bag];
  v8f acc = {};
  for (int t = 0; t < len; t += 2) {
    int n1   = indices[start + t];
    int has2 = (t + 1 < len);
    int n2   = has2 ? indices[start + t + 1] : n1;
    int i3a = n1 % TT_P2, i2a = (n1 / TT_P2) % TT_P1, i1a = n1 / (TT_P1 * TT_P2);
    int i3b = n2 % TT_P2, i2b = (n2 / TT_P2) % TT_P1, i1b = n2 / (TT_P1 * TT_P2);
    const float* ab1 = AB + (size_t)(i1a * TT_P1 + i2a) * AB_ROW;
    const float* ab2 = AB + (size_t)(i1b * TT_P1 + i2b) * AB_ROW;
    const float* c1  = core2 + (size_t)i3a * 256;    // [r2=32][q3=8]
    const float* c2  = core2 + (size_t)i3b * 256;
#pragma unroll
    for (int kk = 0; kk < 16; ++kk) {                // K = 64 in chunks of 4
      const float* ab = (kk < 8) ? ab1 : ab2;
      int kl = (kk * 4 + hi2) & 31;                  // local k within one index
      v2f A;  A.x = ab[ml * 32 + kl];  A.y = ab[ml * 32 + kl + 1];
      v2f Bv; Bv.x = 0.f; Bv.y = 0.f;
      if (kk < 8) {
        if (ml < 8) { Bv.x = c1[kl * 8 + ml];       Bv.y = c1[(kl + 1) * 8 + ml]; }
      } else {
        if ((ml >= 8) && has2) {
          Bv.x = c2[kl * 8 + (ml - 8)]; Bv.y = c2[(kl + 1) * 8 + (ml - 8)];
        }
      }
      acc = __builtin_amdgcn_wmma_f32_16x16x4_f32(false, A, false, Bv,
                                                  (short)0, acc, false, false);
    }
  }
  // fold the two 8-column groups (slot1 + slot2) and store 128 floats
#pragma unroll
  for (int v = 0; v < 8; ++v) {
    float s = acc[v] + __shfl_xor(acc[v], 8, 32);
    if ((lane & 8) == 0) {
      int M = v + ((lane >> 4) << 3);
      out[(size_t)bag * TT_D + M * 8 + (lane & 7)] = s;
    }
  }
}

// ---------------- Fallback (only if workspace can't hold pair table) -------
__global__ void __launch_bounds__(128)
tt_fallback(const int* __restrict__ indices,
            const int* __restrict__ offsets,
            const float* __restrict__ core0,
            const float* __restrict__ core1,
            const float* __restrict__ core2,
            float* __restrict__ out, int B) {
  __shared__ float ab[16 * 32];
  __shared__ int sbag;
  int n   = blockIdx.x;
  int idx = indices[n];
  int i3 = idx % TT_P2, i2 = (idx / TT_P2) % TT_P1, i1 = idx / (TT_P1 * TT_P2);
  int t  = threadIdx.x;
  const float* a = core0 + (size_t)i1 * 128;
  const float* b = core1 + (size_t)i2 * 4096;
  const float* c = core2 + (size_t)i3 * 256;
  for (int e = t; e < 512; e += 128) {
    int m = e >> 5, s = e & 31;
    int q1 = m >> 2, q2 = m & 3;
    float sum = 0.f;
    for (int r = 0; r < 32; ++r)
      sum += a[q1 * 32 + r] * b[r * 128 + q2 * 32 + s];
    ab[e] = sum;
  }
  if (t == 0) {
    int lo = 0, hi = B - 1;
    while (lo < hi) {
      int mid = (lo + hi + 1) >> 1;
      if (offsets[mid] <= n) lo = mid; else hi = mid - 1;
    }
    sbag = lo;
  }
  __syncthreads();
  int m = t >> 3, q = t & 7;
  float sum = 0.f;
  for (int s = 0; s < 32; ++s) sum += ab[m * 32 + s] * c[s * 8 + q];
  atomicAdd(&out[(size_t)sbag * TT_D + t], sum);
}

// ---------------- launcher ----------------
extern "C" void kernel_launch(void* const* d_in, const int* in_sizes, int n_in,
                              void* d_out, int out_size, void* d_ws, size_t ws_size,
                              hipStream_t stream) {
  const int*   indices = (const int*)d_in[0];
  const int*   lengths = (const int*)d_in[1];
  const float* core0   = (const float*)d_in[2];
  const float* core1   = (const float*)d_in[3];
  const float* core2   = (const float*)d_in[4];
  float*       out     = (float*)d_out;
  int NNZ = in_sizes[0];
  int B   = in_sizes[1];
  if (B <= 0 || NNZ <= 0) return;

  int*   offsets  = (int*)d_ws;
  size_t offbytes = (((size_t)B * sizeof(int)) + 255) & ~(size_t)255;
  float* ABt      = (float*)((char*)d_ws + offbytes);
  size_t need     = offbytes + (size_t)NPAIR * AB_ROW * sizeof(float);  // ~20.5 MB

  tt_scan_offsets<<<1, 256, 0, stream>>>(lengths, offsets, B);

  if (ws_size >= need) {
    tt_phaseA<<<dim3((TT_P0 / 4) * TT_P1), 256, 0, stream>>>(core0, core1, ABt);
    tt_phaseB<<<dim3((B + 7) / 8), 256, 0, stream>>>(indices, lengths, offsets,
                                                     ABt, core2, out, B);
  } else {
    hipMemsetAsync(d_out, 0, (size_t)out_size * sizeof(float), stream);
    tt_fallback<<<dim3(NNZ), 128, 0, stream>>>(indices, offsets,
                                               core0, core1, core2, out, B);
  }
}